// DENIS_78847009620092
// MI455X (gfx1250) — compile-verified
//
#include <hip/hip_runtime.h>
#include <hip/hip_bf16.h>

// ---------------------------------------------------------------------------
// Problem constants (from reference)
// ---------------------------------------------------------------------------
#define BATCH 4096
#define TSTEP 64
#define DIMX  128      // input dim
#define H1    256      // enc hidden 1 / aux hidden
#define H2    256      // enc hidden 2
#define EOUT  64       // enc output
#define LDIM  192      // DIMX + EOUT = aux latent dim
#define KO_N  (LDIM*LDIM)   // 36864

typedef __attribute__((ext_vector_type(2))) float v2f;
typedef __attribute__((ext_vector_type(8))) float v8f;

// ---------------------------------------------------------------------------
// WMMA f32 16x16x4 fragment layouts (ISA 7.12.2):
//   A 16x4: lanes 0-15 -> M=lane, K={0,1}; lanes 16-31 -> M=lane-16, K={2,3}
//   B 4x16: lanes 0-15 -> N=lane, K={0,1}; lanes 16-31 -> N=lane-16, K={2,3}
//   C/D:    vgpr r, lanes 0-15 -> (M=r, N=lane); lanes 16-31 -> (M=r+8, N=lane-16)
// We compute C = A(16xK) * W^T with W stored (N x K) row-major, so both A and
// B fragments are contiguous 8-byte loads per lane.
// ---------------------------------------------------------------------------
__device__ __forceinline__ v8f wmma_step(v2f a, v2f b, v8f c)
{
    return __builtin_amdgcn_wmma_f32_16x16x4_f32(false, a, false, b,
                                                 (short)0, c, false, false);
}

// 1x4 N-blocked tile: one A-frag load feeds 4 WMMAs per k-step.
// Computes acc[j] (+)= A(16xK) * W[n0+16j .. n0+16j+15]^T for j=0..3.
__device__ __forceinline__ void wmma_tile_1x4(const float* __restrict__ A, int lda,
                                              const float* __restrict__ W, int ldw,
                                              int K, v8f acc[4])
{
    const int lane = threadIdx.x & 31;
    const int l    = lane & 15;
    const int half = lane >> 4;
    const float* aP = A + (size_t)l * lda + 2 * half;
    const float* wP = W + (size_t)l * ldw + 2 * half;
    #pragma unroll 2
    for (int k0 = 0; k0 < K; k0 += 4) {
        v2f a  = *(const v2f*)(aP + k0);
        v2f b0 = *(const v2f*)(wP + k0);
        v2f b1 = *(const v2f*)(wP + (size_t)16 * ldw + k0);
        v2f b2 = *(const v2f*)(wP + (size_t)32 * ldw + k0);
        v2f b3 = *(const v2f*)(wP + (size_t)48 * ldw + k0);
        acc[0] = wmma_step(a, b0, acc[0]);
        acc[1] = wmma_step(a, b1, acc[1]);
        acc[2] = wmma_step(a, b2, acc[2]);
        acc[3] = wmma_step(a, b3, acc[3]);
    }
}

__device__ __forceinline__ void store_tile(float* __restrict__ C, int ldc, v8f c)
{
    const int lane = threadIdx.x & 31;
    const int l    = lane & 15;
    const int half = lane >> 4;
    #pragma unroll
    for (int r = 0; r < 8; ++r)
        C[(size_t)(r + 8 * half) * ldc + l] = c[r];
}

// ---------------------------------------------------------------------------
// Kernel 1: h = relu(x[:,0,:] @ aux_w0^T + aux_b0)      (4096 x 256)
// One 16x64 strip per wave: 256 M-tiles x 4 N-groups = 1024 waves.
// ---------------------------------------------------------------------------
__global__ __launch_bounds__(256)
void aux_hidden_kernel(const float* __restrict__ x,     // (B,T,DIMX)
                       const float* __restrict__ w0,    // (H1, DIMX)
                       const float* __restrict__ b0,    // (H1)
                       float* __restrict__ h)           // (B, H1)
{
    const int wave = (blockIdx.x * blockDim.x + threadIdx.x) >> 5;
    const int mt = wave >> 2;                 // 0..255
    const int ng = wave & 3;                  // 0..3 (64-wide N group)
    const int lane = threadIdx.x & 31;
    const int l = lane & 15;

    const float* A = x + (size_t)(mt * 16) * (TSTEP * DIMX);   // x0 rows, stride T*DIMX
    const float* W = w0 + (size_t)(ng * 64) * DIMX;

    v8f acc[4] = {};
    wmma_tile_1x4(A, TSTEP * DIMX, W, DIMX, DIMX, acc);

    #pragma unroll
    for (int j = 0; j < 4; ++j) {
        const float bias = b0[ng * 64 + j * 16 + l];
        #pragma unroll
        for (int r = 0; r < 8; ++r) {
            float v = acc[j][r] + bias;
            acc[j][r] = v > 0.f ? v : 0.f;
        }
        store_tile(h + (size_t)(mt * 16) * H1 + ng * 64 + j * 16, H1, acc[j]);
    }
}

// ---------------------------------------------------------------------------
// Kernel 2: ko = h @ aux_w1^T     (4096 x 36864), K = 256
// 4x4 register-blocked: each wave computes a 64x64 tile (16 v8f accumulators),
// 4 A-frags + 4 B-frags -> 16 WMMAs per k-step (4x L2 traffic reduction).
// 64 x 576 macro tiles = 36864 waves -> 4608 blocks x 256 threads.
// ---------------------------------------------------------------------------
__global__ __launch_bounds__(256)
void ko_gemm_kernel(const float* __restrict__ h,    // (B, H1)
                    const float* __restrict__ w1,   // (KO_N, H1)
                    float* __restrict__ ko)         // (B, KO_N)
{
    const int wave = (blockIdx.x * blockDim.x + threadIdx.x) >> 5;
    const int NT = KO_N / 64;                 // 576
    const int mt = wave / NT;                 // 0..63
    const int nt = wave - mt * NT;            // 0..575
    const int lane = threadIdx.x & 31;
    const int l = lane & 15, half = lane >> 4;

    const float* aBase = h  + (size_t)(mt * 64 + l) * H1 + 2 * half;
    const float* bBase = w1 + (size_t)(nt * 64 + l) * H1 + 2 * half;

    v8f acc[4][4] = {};
    #pragma unroll 2
    for (int k0 = 0; k0 < H1; k0 += 4) {
        v2f a[4], b[4];
        #pragma unroll
        for (int i = 0; i < 4; ++i) {
            a[i] = *(const v2f*)(aBase + (size_t)(i * 16) * H1 + k0);
            b[i] = *(const v2f*)(bBase + (size_t)(i * 16) * H1 + k0);
        }
        #pragma unroll
        for (int i = 0; i < 4; ++i)
            #pragma unroll
            for (int j = 0; j < 4; ++j)
                acc[i][j] = wmma_step(a[i], b[j], acc[i][j]);
    }

    float* cBase = ko + (size_t)(mt * 64) * KO_N + (size_t)(nt * 64);
    #pragma unroll
    for (int i = 0; i < 4; ++i)
        #pragma unroll
        for (int j = 0; j < 4; ++j)
            store_tile(cBase + (size_t)(i * 16) * KO_N + j * 16, KO_N, acc[i][j]);
}

// ---------------------------------------------------------------------------
// Kernel 3: fused 3-layer encoder MLP + concat with x -> enc_gt (262144 x 192)
// One 16-row tile per wave; per-wave LDS: e1(16x256) + e2(16x256) = 32 KB.
// 4 waves/block -> 128 KB dynamic LDS (WGP has 320 KB).
// ---------------------------------------------------------------------------
__global__ __launch_bounds__(128)
void enc_fused_kernel(const float* __restrict__ x,    // (262144, DIMX) flat
                      const float* __restrict__ w0, const float* __restrict__ b0,
                      const float* __restrict__ w1, const float* __restrict__ b1,
                      const float* __restrict__ w2,  // (EOUT, H2)
                      float* __restrict__ enc_gt)    // (262144, LDIM)
{
    extern __shared__ float smem[];
    const int waveInBlk = threadIdx.x >> 5;
    float* e1 = smem + waveInBlk * 8192;   // 16 x 256
    float* e2 = e1 + 4096;                 // 16 x 256

    const int tile = blockIdx.x * 4 + waveInBlk;   // 0..16383
    const int row0 = tile * 16;
    const int lane = threadIdx.x & 31;
    const int l = lane & 15, half = lane >> 4;

    const float* A0 = x + (size_t)row0 * DIMX;

    // Layer 1: 128 -> 256, relu
    #pragma unroll 1
    for (int ng = 0; ng < H1 / 64; ++ng) {
        v8f acc[4] = {};
        wmma_tile_1x4(A0, DIMX, w0 + (size_t)(ng * 64) * DIMX, DIMX, DIMX, acc);
        #pragma unroll
        for (int j = 0; j < 4; ++j) {
            const float bias = b0[ng * 64 + j * 16 + l];
            #pragma unroll
            for (int r = 0; r < 8; ++r) {
                float v = acc[j][r] + bias;
                e1[(r + 8 * half) * H1 + ng * 64 + j * 16 + l] = v > 0.f ? v : 0.f;
            }
        }
    }

    // Layer 2: 256 -> 256, relu  (A from LDS)
    #pragma unroll 1
    for (int ng = 0; ng < H2 / 64; ++ng) {
        v8f acc[4] = {};
        wmma_tile_1x4(e1, H1, w1 + (size_t)(ng * 64) * H1, H1, H1, acc);
        #pragma unroll
        for (int j = 0; j < 4; ++j) {
            const float bias = b1[ng * 64 + j * 16 + l];
            #pragma unroll
            for (int r = 0; r < 8; ++r) {
                float v = acc[j][r] + bias;
                e2[(r + 8 * half) * H2 + ng * 64 + j * 16 + l] = v > 0.f ? v : 0.f;
            }
        }
    }

    // Layer 3: 256 -> 64, no bias; write into enc_gt[..., 128:192]
    float* outRow = enc_gt + (size_t)row0 * LDIM;
    {
        v8f acc[4] = {};
        wmma_tile_1x4(e2, H2, w2, H2, H2, acc);
        #pragma unroll
        for (int j = 0; j < 4; ++j)
            #pragma unroll
            for (int r = 0; r < 8; ++r)
                outRow[(size_t)(r + 8 * half) * LDIM + DIMX + j * 16 + l] = acc[j][r];
    }

    // Copy x portion into enc_gt[..., 0:128]
    for (int r = 0; r < 16; ++r)
        for (int col = lane; col < DIMX; col += 32)
            outRow[(size_t)r * LDIM + col] = x[(size_t)(row0 + r) * DIMX + col];
}

// ---------------------------------------------------------------------------
// Kernel 4: recurrence  v_{t+1} = v_t @ ko[b]   (serial in t, parallel in b)
// One block per batch; ko[b] (147 KB) brought into LDS with the CDNA5
// async-to-LDS path (global_load_async_to_lds_b128, ASYNCcnt), then 192
// threads do the ping-pong vec x mat chain out of LDS.
// ---------------------------------------------------------------------------
__global__ __launch_bounds__(192)
void scan_kernel(const float* __restrict__ enc_gt,  // (B, T, LDIM)
                 const float* __restrict__ ko,      // (B, LDIM, LDIM)
                 float* __restrict__ traj)          // (B, T, LDIM)
{
    extern __shared__ float s[];
    float* kos = s;                 // LDIM*LDIM
    float* va  = s + KO_N;          // LDIM
    float* vb  = va + LDIM;         // LDIM

    const int b   = blockIdx.x;
    const int tid = threadIdx.x;    // 0..191

    // Async DMA ko[b] -> LDS: each lane moves 16 B per issue, 48 issues cover
    // 36864 floats. LDS dest offset = low 32 bits of the generic shared ptr
    // (LDS aperture keeps the workgroup-relative offset in addr[31:0]).
    {
        const float* src0 = ko + (size_t)b * KO_N + (size_t)tid * 4;
        unsigned dst0 = (unsigned)(uintptr_t)(kos + (size_t)tid * 4);
        #pragma unroll 4
        for (int it = 0; it < KO_N / (LDIM * 4); ++it) {      // 48 iterations
            unsigned long long g = (unsigned long long)(uintptr_t)src0
                                 + (unsigned long long)it * (LDIM * 16);
            unsigned d = dst0 + (unsigned)(it * (LDIM * 16));
            asm volatile("global_load_async_to_lds_b128 %0, %1, off"
                         :: "v"(d), "v"(g) : "memory");
        }
        asm volatile("s_wait_asynccnt 0" ::: "memory");
    }

    const float v0 = enc_gt[(size_t)b * TSTEP * LDIM + tid];
    va[tid] = v0;
    float* trow = traj + (size_t)b * TSTEP * LDIM;
    trow[tid] = v0;
    __syncthreads();   // also publishes every wave's async LDS fill

    float* cur = va;
    float* nxt = vb;
    for (int t = 1; t < TSTEP; ++t) {
        float acc = 0.f;
        #pragma unroll 8
        for (int i = 0; i < LDIM; ++i)
            acc = fmaf(cur[i], kos[i * LDIM + tid], acc);
        nxt[tid] = acc;
        trow[(size_t)t * LDIM + tid] = acc;
        __syncthreads();
        float* tmp = cur; cur = nxt; nxt = tmp;
    }
}

// ---------------------------------------------------------------------------
// Launch: inputs order = x, enc_w0, enc_b0, enc_w1, enc_b1, enc_w2,
//                        aux_w0, aux_b0, aux_w1
// d_out = [enc_gt | enc_traj | ko] flat f32
// ---------------------------------------------------------------------------
extern "C" void kernel_launch(void* const* d_in, const int* in_sizes, int n_in,
                              void* d_out, int out_size, void* d_ws, size_t ws_size,
                              hipStream_t stream)
{
    const float* x      = (const float*)d_in[0];
    const float* enc_w0 = (const float*)d_in[1];
    const float* enc_b0 = (const float*)d_in[2];
    const float* enc_w1 = (const float*)d_in[3];
    const float* enc_b1 = (const float*)d_in[4];
    const float* enc_w2 = (const float*)d_in[5];
    const float* aux_w0 = (const float*)d_in[6];
    const float* aux_b0 = (const float*)d_in[7];
    const float* aux_w1 = (const float*)d_in[8];

    float* out      = (float*)d_out;
    const size_t enc_gt_elems = (size_t)BATCH * TSTEP * LDIM;   // 50,331,648
    float* enc_gt   = out;
    float* enc_traj = out + enc_gt_elems;
    float* ko       = out + 2 * enc_gt_elems;

    float* h = (float*)d_ws;                                     // 4096 x 256

    // 1) aux hidden: 1024 waves -> 128 blocks x 256 threads
    aux_hidden_kernel<<<128, 256, 0, stream>>>(x, aux_w0, aux_b0, h);

    // 2) ko GEMM: 36864 waves (64x64 tiles) -> 4608 blocks x 256 threads
    ko_gemm_kernel<<<4608, 256, 0, stream>>>(h, aux_w1, ko);

    // 3) fused encoder: 16384 tiles / 4 waves per block -> 4096 blocks
    enc_fused_kernel<<<4096, 128, 4 * 8192 * sizeof(float), stream>>>(
        x, enc_w0, enc_b0, enc_w1, enc_b1, enc_w2, enc_gt);

    // 4) scan: one block per batch, ko[b] async-DMA'd into LDS
    scan_kernel<<<BATCH, LDIM, (KO_N + 2 * LDIM) * sizeof(float), stream>>>(
        enc_gt, ko, enc_traj);
}